// GIN_78065325572476
// MI455X (gfx1250) — compile-verified
//
#include <hip/hip_runtime.h>
#include <hip/hip_bf16.h>

#define NNODES  100000
#define NEDGES  1200000
#define DH      64
#define DOUT    16
#define NGIN    4
#define NPRED   5
#define NGRAPHS 512
#define BN_EPS  1e-5f

typedef __attribute__((ext_vector_type(2))) float v2f;
typedef __attribute__((ext_vector_type(8))) float v8f;

// ---------------------------------------------------------------- utilities

__global__ void zero_kernel(float* __restrict__ p, int n) {
    int i = blockIdx.x * blockDim.x + threadIdx.x;
    int stride = gridDim.x * blockDim.x;
    for (; i < n; i += stride) p[i] = 0.0f;
}

// --------------------------------------------------- edge gather/scatter-add
// tid -> (edge, 16B feature chunk). Lanes 0-15 of a wave cover one edge's
// full 64-float row (coalesced 256B gather); atomics land in L2 (agg fits).
__global__ void scatter_kernel(const float* __restrict__ h,
                               const int* __restrict__ src,
                               const int* __restrict__ dst,
                               float* __restrict__ agg) {
    int tid = blockIdx.x * blockDim.x + threadIdx.x;
    if (tid >= NEDGES * 16) return;
    int e = tid >> 4;
    int c = (tid & 15) << 2;
    int s = src[e], d = dst[e];
    const float4 v = *reinterpret_cast<const float4*>(h + (size_t)s * DH + c);
    float* o = agg + (size_t)d * DH + c;
    atomicAdd(o + 0, v.x);
    atomicAdd(o + 1, v.y);
    atomicAdd(o + 2, v.z);
    atomicAdd(o + 3, v.w);
}

// ---------------------------------------------------------- WMMA tile core
// One wave computes a 16x16 output tile of (Atile[16x64] @ B[64xldb]) + bias.
// Atile lives in LDS, padded to 65 floats/row (conflict-free frag reads).
// Fragment layout per CDNA5 ISA 7.12.2 (fp32 16x16x4).
__device__ __forceinline__ v8f wmma_row_tile(const float* __restrict__ Atile,
                                             const float* __restrict__ B,
                                             int ldb, int ncol, float bias,
                                             int half, int mn) {
    v8f acc;
#pragma unroll
    for (int r = 0; r < 8; ++r) acc[r] = bias;
#pragma unroll
    for (int k0 = 0; k0 < DH; k0 += 4) {
        int k = k0 + 2 * half;
        v2f a, b;
        a[0] = Atile[mn * 65 + k];
        a[1] = Atile[mn * 65 + k + 1];
        b[0] = B[k * ldb + ncol];
        b[1] = B[(k + 1) * ldb + ncol];
        acc = __builtin_amdgcn_wmma_f32_16x16x4_f32(
            false, a, false, b, (short)0, acc, false, false);
    }
    return acc;
}

// --------------------------------------------- GEMM1: pre = (1+eps)h + agg
// T1 = pre @ W1 + b1 ; accumulate column sum / sumsq for BN1.
// block = 128 threads (4 waves), blockIdx = 16-row tile, wave = 16-col tile.
__global__ void gemm1_kernel(const float* __restrict__ h,
                             const float* __restrict__ agg,
                             const float* __restrict__ eps_arr, int layer,
                             const float* __restrict__ W1,   // [64][64] (k,n)
                             const float* __restrict__ b1,   // [64]
                             float* __restrict__ T1,
                             float* __restrict__ stats) {    // [0:64]sum [64:128]sq
    __shared__ float Atile[16 * 65];
    __shared__ float ssum[DH], ssq[DH];
    const int tid = threadIdx.x;
    const int row0 = blockIdx.x * 16;
    const float ep = 1.0f + eps_arr[layer];
    if (tid < DH) { ssum[tid] = 0.0f; ssq[tid] = 0.0f; }
    for (int idx = tid; idx < 16 * DH; idx += blockDim.x) {
        int r = idx >> 6, c = idx & 63;
        size_t g = (size_t)(row0 + r) * DH + c;
        Atile[r * 65 + c] = ep * h[g] + agg[g];
    }
    __syncthreads();
    const int wave = tid >> 5;
    const int lane = tid & 31;
    const int half = lane >> 4;
    const int mn   = lane & 15;
    const int ncol = wave * 16 + mn;
    v8f acc = wmma_row_tile(Atile, W1, DH, ncol, b1[ncol], half, mn);
    float psum = 0.0f, psq = 0.0f;
#pragma unroll
    for (int r = 0; r < 8; ++r) {
        float v = acc[r];
        T1[(size_t)(row0 + r + 8 * half) * DH + ncol] = v;
        psum += v; psq += v * v;
    }
    atomicAdd(&ssum[ncol], psum);
    atomicAdd(&ssq[ncol], psq);
    __syncthreads();
    if (tid < DH)            atomicAdd(&stats[tid], ssum[tid]);
    else if (tid < 2 * DH)   atomicAdd(&stats[tid], ssq[tid - DH]);
}

// ------------------------------- GEMM2: T2 = relu(bn1(T1)) @ W2 + b2 + stats2
__global__ void gemm2_kernel(const float* __restrict__ T1,
                             const float* __restrict__ stats1,
                             const float* __restrict__ g1,
                             const float* __restrict__ be1,
                             const float* __restrict__ W2,
                             const float* __restrict__ b2,
                             float* __restrict__ T2,
                             float* __restrict__ stats2) {
    __shared__ float Atile[16 * 65];
    __shared__ float ssum[DH], ssq[DH];
    const int tid = threadIdx.x;
    const int row0 = blockIdx.x * 16;
    const float invN = 1.0f / (float)NNODES;
    if (tid < DH) { ssum[tid] = 0.0f; ssq[tid] = 0.0f; }
    for (int idx = tid; idx < 16 * DH; idx += blockDim.x) {
        int r = idx >> 6, c = idx & 63;
        float m   = stats1[c] * invN;
        float var = stats1[DH + c] * invN - m * m;
        float rs  = rsqrtf(var + BN_EPS);
        float x   = T1[(size_t)(row0 + r) * DH + c];
        float xn  = (x - m) * rs * g1[c] + be1[c];
        Atile[r * 65 + c] = fmaxf(xn, 0.0f);
    }
    __syncthreads();
    const int wave = tid >> 5;
    const int lane = tid & 31;
    const int half = lane >> 4;
    const int mn   = lane & 15;
    const int ncol = wave * 16 + mn;
    v8f acc = wmma_row_tile(Atile, W2, DH, ncol, b2[ncol], half, mn);
    float psum = 0.0f, psq = 0.0f;
#pragma unroll
    for (int r = 0; r < 8; ++r) {
        float v = acc[r];
        T2[(size_t)(row0 + r + 8 * half) * DH + ncol] = v;
        psum += v; psq += v * v;
    }
    atomicAdd(&ssum[ncol], psum);
    atomicAdd(&ssq[ncol], psq);
    __syncthreads();
    if (tid < DH)            atomicAdd(&stats2[tid], ssum[tid]);
    else if (tid < 2 * DH)   atomicAdd(&stats2[tid], ssq[tid - DH]);
}

// ------------------------ h' = relu(bn2(T2)) ; fused sum-pool into pooled[l+1]
__global__ void bnrelu_pool_kernel(const float* __restrict__ T2,
                                   const float* __restrict__ stats2,
                                   const float* __restrict__ g2,
                                   const float* __restrict__ be2,
                                   const int* __restrict__ graph_id,
                                   float* __restrict__ hout,
                                   float* __restrict__ pooled) {
    int tid = blockIdx.x * blockDim.x + threadIdx.x;
    if (tid >= NNODES * 16) return;
    int node = tid >> 4;
    int c0 = (tid & 15) << 2;
    int g = graph_id[node];
    const float invN = 1.0f / (float)NNODES;
    float4 x = *reinterpret_cast<const float4*>(T2 + (size_t)node * DH + c0);
    float* po = pooled + (size_t)g * DH + c0;
    float4 o;
#pragma unroll
    for (int j = 0; j < 4; ++j) {
        int c = c0 + j;
        float m   = stats2[c] * invN;
        float var = stats2[DH + c] * invN - m * m;
        float rs  = rsqrtf(var + BN_EPS);
        float xv  = (&x.x)[j];
        float xn  = fmaxf((xv - m) * rs * g2[c] + be2[c], 0.0f);
        (&o.x)[j] = xn;
        atomicAdd(po + j, xn);
    }
    *reinterpret_cast<float4*>(hout + (size_t)node * DH + c0) = o;
}

// ------------------------------------- pooling of the raw input (layer 0 rep)
__global__ void pool_kernel(const float* __restrict__ h,
                            const int* __restrict__ graph_id,
                            float* __restrict__ pooled) {
    int tid = blockIdx.x * blockDim.x + threadIdx.x;
    if (tid >= NNODES * 16) return;
    int node = tid >> 4;
    int c0 = (tid & 15) << 2;
    int g = graph_id[node];
    float4 v = *reinterpret_cast<const float4*>(h + (size_t)node * DH + c0);
    float* po = pooled + (size_t)g * DH + c0;
    atomicAdd(po + 0, v.x);
    atomicAdd(po + 1, v.y);
    atomicAdd(po + 2, v.z);
    atomicAdd(po + 3, v.w);
}

// ------------------- score = sum_i pooled_i @ Wp[i] + sum_i bp[i]  (WMMA f32)
// 512 rows -> 32 row tiles, N=16 is exactly one tile; 8 blocks x 4 waves.
__global__ void final_kernel(const float* __restrict__ pooled,  // [5][512][64]
                             const float* __restrict__ Wp,      // [5][64][16]
                             const float* __restrict__ bp,      // [5][16]
                             float* __restrict__ score) {       // [512][16]
    int wid  = (blockIdx.x * blockDim.x + threadIdx.x) >> 5;
    int lane = threadIdx.x & 31;
    if (wid >= NGRAPHS / 16) return;
    int half = lane >> 4;
    int mn   = lane & 15;
    int row0 = wid * 16;
    float bias = 0.0f;
    for (int i = 0; i < NPRED; ++i) bias += bp[i * DOUT + mn];
    v8f acc;
#pragma unroll
    for (int r = 0; r < 8; ++r) acc[r] = bias;
    for (int i = 0; i < NPRED; ++i) {
        const float* A = pooled + (size_t)i * NGRAPHS * DH;
        const float* B = Wp + (size_t)i * DH * DOUT;
#pragma unroll
        for (int k0 = 0; k0 < DH; k0 += 4) {
            int k = k0 + 2 * half;
            v2f a, b;
            a[0] = A[(size_t)(row0 + mn) * DH + k];
            a[1] = A[(size_t)(row0 + mn) * DH + k + 1];
            b[0] = B[k * DOUT + mn];
            b[1] = B[(k + 1) * DOUT + mn];
            acc = __builtin_amdgcn_wmma_f32_16x16x4_f32(
                false, a, false, b, (short)0, acc, false, false);
        }
    }
#pragma unroll
    for (int r = 0; r < 8; ++r)
        score[(size_t)(row0 + r + 8 * half) * DOUT + mn] = acc[r];
}

// --------------------------------------------------------------------- host

extern "C" void kernel_launch(void* const* d_in, const int* in_sizes, int n_in,
                              void* d_out, int out_size, void* d_ws, size_t ws_size,
                              hipStream_t stream) {
    const float* h_in = (const float*)d_in[0];
    const int*   esrc = (const int*)d_in[1];
    const int*   edst = (const int*)d_in[2];
    const int*   gid  = (const int*)d_in[3];
    const float* eps  = (const float*)d_in[4];
    const float* W1   = (const float*)d_in[5];
    const float* b1   = (const float*)d_in[6];
    const float* g1   = (const float*)d_in[7];
    const float* be1  = (const float*)d_in[8];
    const float* W2   = (const float*)d_in[9];
    const float* b2   = (const float*)d_in[10];
    const float* g2   = (const float*)d_in[11];
    const float* be2  = (const float*)d_in[12];
    const float* Wp   = (const float*)d_in[13];
    const float* bp   = (const float*)d_in[14];
    float* score = (float*)d_out;

    const size_t NF = (size_t)NNODES * DH;
    float* ws     = (float*)d_ws;
    float* hcur   = ws;             // h after each layer
    float* bufA   = ws + NF;        // agg, then reused as T2
    float* bufB   = ws + 2 * NF;    // T1
    float* stats  = ws + 3 * NF;    // 256 floats: sum1,sq1,sum2,sq2
    float* pooled = stats + 256;    // 5 * 512 * 64

    const int THREADS = 256;
    const int nodeGrid = (NNODES * 16 + THREADS - 1) / THREADS;
    const int edgeGrid = (NEDGES * 16 + THREADS - 1) / THREADS;

    zero_kernel<<<256, THREADS, 0, stream>>>(pooled, NPRED * NGRAPHS * DH);
    pool_kernel<<<nodeGrid, THREADS, 0, stream>>>(h_in, gid, pooled);

    const float* hsrc = h_in;
    for (int i = 0; i < NGIN; ++i) {
        zero_kernel<<<1024, THREADS, 0, stream>>>(bufA, (int)NF);
        zero_kernel<<<1, THREADS, 0, stream>>>(stats, 256);
        scatter_kernel<<<edgeGrid, THREADS, 0, stream>>>(hsrc, esrc, edst, bufA);
        gemm1_kernel<<<NNODES / 16, 128, 0, stream>>>(
            hsrc, bufA, eps, i,
            W1 + (size_t)i * DH * DH, b1 + (size_t)i * DH, bufB, stats);
        gemm2_kernel<<<NNODES / 16, 128, 0, stream>>>(
            bufB, stats, g1 + (size_t)i * DH, be1 + (size_t)i * DH,
            W2 + (size_t)i * DH * DH, b2 + (size_t)i * DH, bufA, stats + 128);
        bnrelu_pool_kernel<<<nodeGrid, THREADS, 0, stream>>>(
            bufA, stats + 128, g2 + (size_t)i * DH, be2 + (size_t)i * DH,
            gid, hcur, pooled + (size_t)(i + 1) * NGRAPHS * DH);
        hsrc = hcur;
    }
    final_kernel<<<8, 128, 0, stream>>>(pooled, Wp, bp, score);
}